// ARLayer_87282325390073
// MI455X (gfx1250) — compile-verified
//
#include <hip/hip_runtime.h>

typedef __attribute__((ext_vector_type(2))) float v2f;
typedef __attribute__((ext_vector_type(8))) float v8f;

#define N_ENT      1000000
#define N_REL      1000
#define DIM        128
#define R_TYPES    8
#define E_PER_TYPE 131072
#define SCORE_DIM  150000
#define OUT_TOTAL  (R_TYPES * SCORE_DIM)

// ---------------------------------------------------------------------------
// Kernel 1: rowsum of ent_table via V_WMMA_F32_16X16X4_F32 with a ones B.
// One wave32 per 16-row tile. A-fragment layout (ISA 7.12.2, 32-bit A 16x4):
//   lanes 0-15 : M = lane,    VGPR0 = K0, VGPR1 = K1
//   lanes 16-31: M = lane-16, VGPR0 = K2, VGPR1 = K3
// With B = all-ones, every column of D equals the row sum; D layout puts
// rows 0-7 in lane 0's 8 accumulator VGPRs and rows 8-15 in lane 16's.
// ---------------------------------------------------------------------------
__global__ void __launch_bounds__(256)
rowsum_wmma_kernel(const float* __restrict__ ent,
                   float* __restrict__ rowsum,
                   int n_tiles) {
    const int wave = (int)((blockIdx.x * blockDim.x + threadIdx.x) >> 5);
    const int lane = (int)(threadIdx.x & 31);
    if (wave >= n_tiles) return;                 // uniform per wave: EXEC all-1s inside

    const size_t row0  = (size_t)wave * 16;
    const int    m     = lane & 15;              // row within tile
    const int    kbase = (lane >> 4) << 1;       // 0 for low half-wave, 2 for high

    const float* p = ent + (row0 + m) * DIM + kbase;   // 8B-aligned

    const v2f b = {1.0f, 1.0f};                  // ones vector (all B fragments)
    v8f c = {};                                  // f32 accumulator, chained over K

    #pragma unroll
    for (int k = 0; k < DIM; k += 4) {
        v2f a = *(const v2f*)(p + k);            // global_load_b64: 16B/row/step
        c = __builtin_amdgcn_wmma_f32_16x16x4_f32(
                /*neg_a=*/false, a, /*neg_b=*/false, b,
                /*c_mod=*/(short)0, c, /*reuse_a=*/false, /*reuse_b=*/false);
    }

    // Column N=0 lives in lanes 0 (M=0..7) and 16 (M=8..15).
    if ((lane & 15) == 0) {
        float* o = rowsum + row0 + ((size_t)(lane >> 4) << 3);
        #pragma unroll
        for (int j = 0; j < 8; ++j) o[j] = c[j];
    }
}

// ---------------------------------------------------------------------------
// Kernel 2: rel_table rowsums (tiny: 1000 x 128 = 512 KB, one thread per row)
// ---------------------------------------------------------------------------
__global__ void relsum_kernel(const float* __restrict__ rel,
                              float* __restrict__ relsum) {
    const int i = (int)(blockIdx.x * blockDim.x + threadIdx.x);
    if (i >= N_REL) return;
    const float4* p = (const float4*)(rel + (size_t)i * DIM);
    float s = 0.0f;
    #pragma unroll
    for (int j = 0; j < DIM / 4; ++j) {
        float4 v = p[j];
        s += v.x + v.y + v.z + v.w;
    }
    relsum[i] = s;
}

// ---------------------------------------------------------------------------
// Kernel 3: edge scoring + zero-padding. Gathers are 4B hits in a 4MB
// L2-resident table. Padding region is rewritten to 0 every call (d_out is
// poisoned by the harness).
// ---------------------------------------------------------------------------
__global__ void __launch_bounds__(256)
score_kernel(const int* __restrict__ src, const int* __restrict__ dst,
             const int* __restrict__ rix,
             const float* __restrict__ rowsum, const float* __restrict__ relsum,
             float* __restrict__ out) {
    const int i = (int)(blockIdx.x * blockDim.x + threadIdx.x);
    if (i >= OUT_TOTAL) return;
    const int r = i / SCORE_DIM;
    const int e = i - r * SCORE_DIM;
    float v = 0.0f;
    if (e < E_PER_TYPE) {
        const int idx = r * E_PER_TYPE + e;
        v = rowsum[src[idx]] + relsum[rix[idx]] - rowsum[dst[idx]];
    }
    out[i] = v;
}

// ---------------------------------------------------------------------------
// Fallback if workspace is too small: direct per-edge gather-reduce.
// ---------------------------------------------------------------------------
__global__ void __launch_bounds__(256)
score_direct_kernel(const float* __restrict__ ent, const float* __restrict__ rel,
                    const int* __restrict__ src, const int* __restrict__ dst,
                    const int* __restrict__ rix, float* __restrict__ out) {
    const int i = (int)(blockIdx.x * blockDim.x + threadIdx.x);
    if (i >= OUT_TOTAL) return;
    const int r = i / SCORE_DIM;
    const int e = i - r * SCORE_DIM;
    float v = 0.0f;
    if (e < E_PER_TYPE) {
        const int idx = r * E_PER_TYPE + e;
        const float4* h = (const float4*)(ent + (size_t)src[idx] * DIM);
        const float4* t = (const float4*)(ent + (size_t)dst[idx] * DIM);
        const float4* q = (const float4*)(rel + (size_t)rix[idx] * DIM);
        float s = 0.0f;
        #pragma unroll
        for (int j = 0; j < DIM / 4; ++j) {
            float4 a = h[j], b = q[j], c = t[j];
            s += (a.x + b.x - c.x) + (a.y + b.y - c.y) +
                 (a.z + b.z - c.z) + (a.w + b.w - c.w);
        }
        v = s;
    }
    out[i] = v;
}

extern "C" void kernel_launch(void* const* d_in, const int* in_sizes, int n_in,
                              void* d_out, int out_size, void* d_ws, size_t ws_size,
                              hipStream_t stream) {
    const float* ent  = (const float*)d_in[0];
    const float* relt = (const float*)d_in[1];
    const int*   src  = (const int*)d_in[2];
    const int*   dst  = (const int*)d_in[3];
    const int*   rix  = (const int*)d_in[4];
    float*       out  = (float*)d_out;

    const size_t need = (size_t)(N_ENT + N_REL) * sizeof(float);
    if (ws_size >= need) {
        float* rowsum = (float*)d_ws;
        float* relsum = rowsum + N_ENT;

        const int n_tiles = N_ENT / 16;                    // 62500, exact
        const int waves_per_block = 256 / 32;              // 8
        const int blocks1 = (n_tiles + waves_per_block - 1) / waves_per_block;
        rowsum_wmma_kernel<<<blocks1, 256, 0, stream>>>(ent, rowsum, n_tiles);

        relsum_kernel<<<(N_REL + 255) / 256, 256, 0, stream>>>(relt, relsum);

        score_kernel<<<(OUT_TOTAL + 255) / 256, 256, 0, stream>>>(
            src, dst, rix, rowsum, relsum, out);
    } else {
        score_direct_kernel<<<(OUT_TOTAL + 255) / 256, 256, 0, stream>>>(
            ent, relt, src, dst, rix, out);
    }
}